// LoRALayer_14998025797849
// MI455X (gfx1250) — compile-verified
//
#include <hip/hip_runtime.h>

typedef __attribute__((ext_vector_type(2))) float v2f;
typedef __attribute__((ext_vector_type(8))) float v8f;

#define IN_F   4096
#define OUT_F  4096
#define RANK   16
#define SCALING 2.0f

#define M_TILE 16
#define KC     512            // x K-chunk staged in LDS
#define XROW   (KC + 4)       // padded LDS row stride (floats): bank step 4/row
#define NWAVES 8
#define KSLICE (KC / NWAVES)  // 64 K-columns per wave per chunk
#define CSLICE (OUT_F / NWAVES) // 512 output columns per wave

__global__ __launch_bounds__(256)
void lora_fused_wmma(const float* __restrict__ x,
                     const float* __restrict__ lora_A,
                     const float* __restrict__ lora_B,
                     float* __restrict__ out,
                     int rows)
{
    __shared__ float lds_x[M_TILE * XROW];     // 16 x 516 fp32 (~33 KB)
    __shared__ float lds_part[NWAVES * 256];   // per-wave partial h, C-layout (8 KB)
    __shared__ float lds_h[M_TILE * 17];       // reduced h[16][16], stride-17 padded

    const int tid  = threadIdx.x;
    const int wave = tid >> 5;
    const int lane = tid & 31;
    const int m16  = lane & 15;   // M (stage1 A) / N (B frags) index
    const int kh   = lane >> 4;   // selects K pair {0,1} vs {2,3}

    const int row0 = blockIdx.x * M_TILE;
    if (row0 >= rows) return;     // uniform; EXEC stays all-ones for WMMA

    v8f hacc = {};                // partial h accumulator (16x16 f32 C-layout)

    // ---------------- Stage 1: h = x_tile @ lora_A^T ----------------
    for (int kc = 0; kc < IN_F; kc += KC) {
        // Cooperative coalesced load: x[16, KC] -> LDS (float4 per thread)
        for (int i = tid; i < M_TILE * (KC / 4); i += 256) {
            const int r  = i >> 7;        // / (KC/4) == /128
            const int c4 = i & 127;
            float4 v = ((const float4*)(x + (size_t)(row0 + r) * IN_F + kc))[c4];
            float* dst = &lds_x[r * XROW + c4 * 4];
            dst[0] = v.x; dst[1] = v.y; dst[2] = v.z; dst[3] = v.w;
        }
        __syncthreads();

        // Each wave accumulates over its 64-column K-slice of this chunk
        const int kw = wave * KSLICE;
#pragma unroll
        for (int i = 0; i < KSLICE; i += 4) {
            const int klocal = kw + i;          // K within chunk
            const int kglob  = kc + klocal;     // absolute K
            // A fragment (x): lane m16 row, K = klocal + 2*kh + {0,1}
            const float* xa = &lds_x[m16 * XROW + klocal + 2 * kh];
            v2f afrag; afrag[0] = xa[0]; afrag[1] = xa[1];
            // B fragment (A^T): Bt[k][r] = lora_A[r][k]; lane holds rank r = m16
            const float* ba = lora_A + (size_t)m16 * IN_F + kglob + 2 * kh;
            v2f bfrag; bfrag[0] = ba[0]; bfrag[1] = ba[1];
            hacc = __builtin_amdgcn_wmma_f32_16x16x4_f32(
                false, afrag, false, bfrag, (short)0, hacc, false, false);
        }
        __syncthreads();
    }

    // ---------- Cross-wave reduction of partial h into LDS ----------
    {
        float* p = &lds_part[wave * 256];
#pragma unroll
        for (int v = 0; v < 8; ++v) p[v * 32 + lane] = hacc[v];
    }
    __syncthreads();
    {
        // 256 threads: thread t owns h[m][n]; C-layout: M = v + 8*(l>=16), N = l&15
        const int m = tid >> 4, n = tid & 15;
        const int lsrc = ((m >> 3) << 4) + n;
        const int v = m & 7;
        float s = 0.f;
#pragma unroll
        for (int w = 0; w < NWAVES; ++w) s += lds_part[w * 256 + v * 32 + lsrc];
        lds_h[m * 17 + n] = s;
    }
    __syncthreads();

    // ---------------- Stage 2: out_tile = h @ lora_B^T * 2 ----------------
    const int colw = wave * CSLICE;
    for (int ct = 0; ct < CSLICE; ct += 16) {
        const int col0 = colw + ct;
        v8f acc = {};
#pragma unroll
        for (int kk = 0; kk < 4; ++kk) {
            // A fragment from h: lane m16 row, K(rank) = kk*4 + 2*kh + {0,1}
            const float* ha = &lds_h[m16 * 17 + kk * 4 + 2 * kh];
            v2f afrag; afrag[0] = ha[0]; afrag[1] = ha[1];
            // B fragment (B^T): Bt[k][n] = lora_B[col0+n][k]; lane holds n = m16
            const float* bb = lora_B + (size_t)(col0 + m16) * RANK + kk * 4 + 2 * kh;
            v2f bfrag; bfrag[0] = bb[0]; bfrag[1] = bb[1];
            acc = __builtin_amdgcn_wmma_f32_16x16x4_f32(
                false, afrag, false, bfrag, (short)0, acc, false, false);
        }
        // Store: C/D layout -> out[row0 + v + 8*kh][col0 + m16]
        float* o = out + (size_t)row0 * OUT_F + col0 + m16;
#pragma unroll
        for (int v = 0; v < 8; ++v)
            o[(size_t)(v + 8 * kh) * OUT_F] = acc[v] * SCALING;
    }
}

extern "C" void kernel_launch(void* const* d_in, const int* in_sizes, int n_in,
                              void* d_out, int out_size, void* d_ws, size_t ws_size,
                              hipStream_t stream) {
    const float* x      = (const float*)d_in[0];   // [4,2048,4096] fp32
    const float* lora_A = (const float*)d_in[1];   // [16,4096] fp32
    const float* lora_B = (const float*)d_in[2];   // [4096,16] fp32
    float* out          = (float*)d_out;           // [4,2048,4096] fp32

    const int rows = in_sizes[0] / IN_F;           // 8192
    dim3 grid(rows / M_TILE);                      // 512 workgroups
    lora_fused_wmma<<<grid, 256, 0, stream>>>(x, lora_A, lora_B, out, rows);
}